// FSANet_43645457662193
// MI455X (gfx1250) — compile-verified
//
#include <hip/hip_runtime.h>
#include <hip/hip_bf16.h>

typedef __attribute__((ext_vector_type(2))) float v2f;
typedef __attribute__((ext_vector_type(8))) float v8f;

#define B_TOT   4096
#define N_IN    2048
#define D_INP   16
#define TB      16            // batch rows per block (WMMA M)
#define NC      32            // n-chunk per pass iteration
#define GPAD    264           // padded LDS row stride (NC*8=256 -> +8 to break bank conflicts)
#define NCHUNK  (N_IN / NC)   // 64
#define DUMP    (48 * GPAD)   // LDS dump slot for lanes with no valid tile-1 column

__global__ __launch_bounds__(256) void fsanet_fused(
    const float* __restrict__ x, const float* __restrict__ w,
    const float* __restrict__ ws1, const float* __restrict__ bs1,
    const float* __restrict__ wc1, const float* __restrict__ bc1,
    const float* __restrict__ wp1, const float* __restrict__ bp1,
    const float* __restrict__ ws2, const float* __restrict__ bs2,
    const float* __restrict__ wc2, const float* __restrict__ bc2,
    const float* __restrict__ wp2, const float* __restrict__ bp2,
    const float* __restrict__ ws3, const float* __restrict__ bs3,
    const float* __restrict__ wc3, const float* __restrict__ bc3,
    const float* __restrict__ wp3, const float* __restrict__ bp3,
    float* __restrict__ out)
{
  // u_hat chunk: [ (b*3+o) ][ nl*8 + d ], padded row stride, + 64-float dump region
  __shared__ float chunkBuf[48 * GPAD + 64];     // ~51 KB
  __shared__ float Vs  [TB][3][8];               // cumulative routing outputs
  __shared__ float sBuf[TB][24];                 // per-b reduced s
  __shared__ float outB[TB][3][8];               // squashed outputs (caps)

  const int tid  = threadIdx.x;
  const int lane = tid & 31;
  const int wv   = tid >> 5;          // wave id 0..7
  const int mrow = lane & 15;         // A: M (batch row) / B,D: N column
  const int kh   = lane >> 4;         // K half select per ISA A/B layout
  const int b0   = blockIdx.x * TB;

  for (int i = tid; i < TB * 24; i += 256) ((float*)Vs)[i] = 0.f;
  __syncthreads();

  const int tb  = tid >> 4;           // routing phase: local batch row
  const int sub = tid & 15;           // routing phase: n-slice within chunk
  const int o0  = mrow >> 3;          // column tile 0: o in {0,1}
  const int d0  = mrow & 7;
  const bool t1v = (mrow < 8);        // column tile 1: only cols 0..7 valid (o=2)
  const float m1 = t1v ? 1.f : 0.f;   // branchless zero-mask for tile-1 B operand

  for (int pass = 0; pass < 3; ++pass) {
    float sacc[24];
    #pragma unroll
    for (int i = 0; i < 24; ++i) sacc[i] = 0.f;

    for (int ch = 0; ch < NCHUNK; ++ch) {
      __syncthreads();  // chunkBuf reusable; Vs stable
      // ---------- WMMA phase: wave wv computes n_local = wv*4 .. wv*4+3 ----------
      #pragma unroll
      for (int q = 0; q < 4; ++q) {
        const int nl = wv * 4 + q;
        const int n  = ch * NC + nl;
        const float* xrow = x + (((size_t)(b0 + mrow)) * N_IN + n) * D_INP;
        const float* wr0  = w + ((((size_t)o0 * N_IN) + n) * 8 + d0) * D_INP;
        const float* wr1  = w + ((((size_t)2  * N_IN) + n) * 8 + d0) * D_INP;
        v8f acc0 = {}; v8f acc1 = {};
        #pragma unroll
        for (int kk = 0; kk < 4; ++kk) {
          v2f a   = *(const v2f*)(xrow + kk * 4 + kh * 2);   // A 16x4 f32 layout
          v2f bb0 = *(const v2f*)(wr0  + kk * 4 + kh * 2);   // B 4x16: lanes=cols
          v2f bt  = *(const v2f*)(wr1  + kk * 4 + kh * 2);   // always in-bounds
          v2f bb1; bb1.x = bt.x * m1; bb1.y = bt.y * m1;     // branchless column mask
          acc0 = __builtin_amdgcn_wmma_f32_16x16x4_f32(
                   false, a, false, bb0, (short)0, acc0, false, false);
          acc1 = __builtin_amdgcn_wmma_f32_16x16x4_f32(
                   false, a, false, bb1, (short)0, acc1, false, false);
        }
        // speculative prefetch of next chunk's x row (OOB prefetch is dropped by HW)
        __builtin_prefetch(xrow + NC * D_INP, 0, 1);
        // D layout: VGPR r = row r (lanes<16) / row r+8 (lanes>=16); col = mrow
        const int base0 = (kh * 8 * 3 + o0) * GPAD + nl * 8 + d0;
        const int base1 = t1v ? ((kh * 8 * 3 + 2) * GPAD + nl * 8 + mrow) : (DUMP + lane);
        const int step1 = t1v ? (3 * GPAD) : 0;
        #pragma unroll
        for (int r = 0; r < 8; ++r) {
          chunkBuf[base0 + r * 3 * GPAD] = acc0[r];
          chunkBuf[base1 + r * step1]    = acc1[r];   // invalid lanes -> dump slot
        }
      }
      __syncthreads();
      // ---------- routing accumulate: thread (tb,sub) covers nl = 2*sub, 2*sub+1 ----------
      #pragma unroll
      for (int q = 0; q < 2; ++q) {
        const int nl = sub * 2 + q;
        float u[3][8], dot[3];
        #pragma unroll
        for (int o = 0; o < 3; ++o) {
          const float* p = &chunkBuf[(tb * 3 + o) * GPAD + nl * 8];
          float dd = 0.f;
          #pragma unroll
          for (int d = 0; d < 8; ++d) { u[o][d] = p[d]; dd += u[o][d] * Vs[tb][o][d]; }
          dot[o] = dd;
        }
        const float mx = fmaxf(dot[0], fmaxf(dot[1], dot[2]));
        const float e0 = __expf(dot[0] - mx), e1 = __expf(dot[1] - mx), e2 = __expf(dot[2] - mx);
        const float inv = 1.f / (e0 + e1 + e2);
        const float c[3] = { e0 * inv, e1 * inv, e2 * inv };
        #pragma unroll
        for (int o = 0; o < 3; ++o)
          #pragma unroll
          for (int d = 0; d < 8; ++d) sacc[o * 8 + d] += c[o] * u[o][d];
      }
    }
    // ---------- reduce s over 16 threads of each b (stays inside 16-lane halves) ----------
    #pragma unroll
    for (int st = 1; st < 16; st <<= 1)
      #pragma unroll
      for (int i = 0; i < 24; ++i) sacc[i] += __shfl_xor(sacc[i], st, 32);
    if (sub == 0)
      #pragma unroll
      for (int i = 0; i < 24; ++i) sBuf[tb][i] = sacc[i];
    __syncthreads();
    // ---------- squash + cumulative V ----------
    if (tid < 48) {
      const int b = tid / 3, o = tid % 3;
      float sv[8], nrm = 0.f;
      #pragma unroll
      for (int d = 0; d < 8; ++d) { sv[d] = sBuf[b][o * 8 + d]; nrm += sv[d] * sv[d]; }
      const float f = nrm / (1.f + nrm) / (sqrtf(nrm) + 1e-8f);
      #pragma unroll
      for (int d = 0; d < 8; ++d) {
        const float ov = f * sv[d];
        outB[b][o][d] = ov;
        Vs[b][o][d] += ov;
      }
    }
    __syncthreads();
  }

  // ---------- SSR heads: one thread per batch row ----------
  if (tid < TB) {
    const int b = tid;
    const float* WS[3] = { ws1, ws2, ws3 };
    const float* BS[3] = { bs1, bs2, bs3 };
    const float* WC[3] = { wc1, wc2, wc3 };
    const float* BC[3] = { bc1, bc2, bc3 };
    const float* WP[3] = { wp1, wp2, wp3 };
    const float* BP[3] = { bp1, bp2, bp3 };
    float A[3][3], D[3][3];
    for (int s = 0; s < 3; ++s) {
      const float* f = &outB[b][s][0];   // caps[:, s, :]
      for (int c = 0; c < 3; ++c) {
        const float sh = tanhf(f[0] * WS[s][c * 2 + 0] + f[1] * WS[s][c * 2 + 1] + BS[s][c]);
        const float sc = tanhf(f[2] * WC[s][c * 2 + 0] + f[3] * WC[s][c * 2 + 1] + BC[s][c]);
        float p[3];
        for (int j = 0; j < 3; ++j) {
          const int r = c * 3 + j;
          float t = BP[s][r];
          for (int tt = 0; tt < 4; ++tt) t += f[4 + tt] * WP[s][r * 4 + tt];
          p[j] = fmaxf(t, 0.f);
        }
        A[s][c] = (sh - 1.f) * p[0] + sh * p[1] + (sh + 1.f) * p[2];
        D[s][c] = 3.f * (1.f + sc);
      }
    }
    for (int c = 0; c < 3; ++c) {
      const float r1 = A[0][c] / D[0][c];
      const float r2 = A[1][c] / (D[0][c] * D[1][c]);
      const float r3 = A[2][c] / (D[0][c] * D[1][c] * D[2][c]);
      out[(size_t)(b0 + b) * 3 + c] = (r1 + r2 + r3) * 99.f;
    }
  }
}

extern "C" void kernel_launch(void* const* d_in, const int* in_sizes, int n_in,
                              void* d_out, int out_size, void* d_ws, size_t ws_size,
                              hipStream_t stream) {
  (void)in_sizes; (void)n_in; (void)out_size; (void)d_ws; (void)ws_size;
  const float* x = (const float*)d_in[0];
  const float* w = (const float*)d_in[1];
  fsanet_fused<<<B_TOT / TB, 256, 0, stream>>>(
      x, w,
      (const float*)d_in[2],  (const float*)d_in[3],
      (const float*)d_in[4],  (const float*)d_in[5],
      (const float*)d_in[6],  (const float*)d_in[7],
      (const float*)d_in[8],  (const float*)d_in[9],
      (const float*)d_in[10], (const float*)d_in[11],
      (const float*)d_in[12], (const float*)d_in[13],
      (const float*)d_in[14], (const float*)d_in[15],
      (const float*)d_in[16], (const float*)d_in[17],
      (const float*)d_in[18], (const float*)d_in[19],
      (float*)d_out);
}